// DFINETransformerModule_11879879540798
// MI455X (gfx1250) — compile-verified
//
#include <hip/hip_runtime.h>
#include <hip/hip_bf16.h>

// ============================================================================
// D-FINE decoder forward for gfx1250 (MI455X), fp32 end-to-end.
// All linear layers / attention GEMMs use V_WMMA_F32_16X16X4_F32 (fp32 tensor
// path) via __builtin_amdgcn_wmma_f32_16x16x4_f32 so numerics stay close to
// the fp32 reference while the matrix pipe does the FLOPs.
//
// Input flattening assumption (documented for iteration):
//   d_in[0] = target (16,300,256)
//   d_in[1] = ref_points_unact (16,300,4)
//   d_in[2] = memory (16,8500,256)
//   d_in[3:] = params leaves, JAX pytree order (dict keys sorted, lists in
//              order, lin dict {'b','w'} -> b then w, ln tuple -> (gamma,beta)):
//     bbox_head[6][3]:(b,w), box_distance_weight(33),
//     layers[6]: ca.aw(b,w), ca.off(b,w), ff1(b,w), ff2(b,w), gate(b,w),
//                gnorm(g,b), norm1(g,b), norm3(g,b),
//                sa.k(b,w), sa.o(b,w), sa.q(b,w), sa.v(b,w)
//     lqe[6][2]:(b,w), pre_bbox_head[3]:(b,w),
//     query_pos_head[2]:(b,w), score_head[6]:(b,w)
// ============================================================================

typedef float v2f __attribute__((ext_vector_type(2)));
typedef float v8f __attribute__((ext_vector_type(8)));

#define BB      16
#define NQ      300
#define HID     256
#define ROWS    (BB * NQ)          // 4800
#define NH      8
#define HD      32
#define LTOT    8500
#define TP      15                 // total points
#define REGBINS 33                 // REG_MAX + 1
#define NCLS    80
#define DFF     1024

// d_out layout (floats)
#define O_BBOX  0                  // 6 * 19200
#define O_LOGIT 115200             // 6 * 384000
#define O_CORN  2419200            // 6 * 633600
#define O_REFS  6220800            // 6 * 19200
#define O_PREB  6336000            // 19200
#define O_PRES  6355200            // 384000

// ---------------------------------------------------------------------------
// Generic batched WMMA fp32 GEMM:  C[z] = act( A[z] * B[z] + bias + addIn )
//   A: M x K, element (m,k) at A + m*lda + k      (lda always even here)
//   B: K x N, element (k,n) at B + k*ldbK + n*ldbN
//   C: M x N, element (m,n) at C + m*ldc + n      (addIn shares C layout)
//   batch index z: zo = z/inner, zi = z%inner; base += zo*s1 + zi*s2
//   act: 0 none, 1 relu, 2 sigmoid, 3 clip(-10,10)
//
// One wave computes NT 16x16 tiles side by side (one A fragment feeds NT
// WMMAs per K step). Out-of-range lanes CLAMP their row/col index instead of
// predicating loads: a clamped A row / B column only corrupts D rows/cols
// that are masked at store time, so the inner loop has zero branches and the
// WMMA runs with full EXEC as required.
// The K loop is 2-stage software pipelined: fragments for step k+4 are issued
// before the WMMAs of step k consume their data, so the compiler can wait on
// loadcnt <= (NT+1) instead of 0 and overlap prefetch with math.
// BMODE==1: ldbK==1 (contiguous K) -> B fragment is an aligned v2f load.
// ---------------------------------------------------------------------------
template <int NT, int BMODE>
__global__ void gemm_wmma_f32(
    const float* __restrict__ A, long long sA1, long long sA2, int lda,
    const float* __restrict__ Bm, long long sB1, long long sB2, int ldbK, int ldbN,
    const float* __restrict__ bias,
    const float* __restrict__ addIn,
    float* __restrict__ C, long long sC1, long long sC2, int ldc,
    int M, int N, int K, int inner, int act)
{
    int z  = blockIdx.z;
    int zo = z / inner, zi = z % inner;
    A  += zo * sA1 + zi * sA2;
    Bm += zo * sB1 + zi * sB2;
    C  += zo * sC1 + zi * sC2;
    if (addIn) addIn += zo * sC1 + zi * sC2;

    int lane = threadIdx.x;        // 0..31, wave32
    int half = lane >> 4;          // 0: lanes 0-15, 1: lanes 16-31
    int l    = lane & 15;
    int m0   = blockIdx.y * 16;
    int n0   = blockIdx.x * (16 * NT);

    // A fragment: lanes 0-15 & 16-31 both hold M=0..15; VGPR j holds K = 2*half+j
    int rowA = m0 + l;
    int rc   = rowA < M ? rowA : M - 1;                  // clamp (see note above)
    const float* Ab = A + (long long)rc * lda + 2 * half;

    const float* Bb[NT];
    #pragma unroll
    for (int t = 0; t < NT; ++t) {
        int col = n0 + t * 16 + l;
        int cc  = col < N ? col : N - 1;                 // clamp
        Bb[t] = Bm + (long long)cc * ldbN;
    }

    v8f acc[NT];
    #pragma unroll
    for (int t = 0; t < NT; ++t) acc[t] = (v8f){};

    // ---- software-pipelined K loop (step 4) ----
    int ka0 = 2 * half;                                  // this half-wave: K = ka, ka+1
    v2f a_cur = *(const v2f*)(Ab + 0);
    v2f b_cur[NT];
    #pragma unroll
    for (int t = 0; t < NT; ++t) {
        if (BMODE == 1) b_cur[t] = *(const v2f*)(Bb[t] + ka0);
        else { b_cur[t][0] = Bb[t][(long long)ka0 * ldbK];
               b_cur[t][1] = Bb[t][(long long)(ka0 + 1) * ldbK]; }
    }

    for (int k0 = 0; k0 < K; k0 += 4) {
        int k1 = (k0 + 4 < K) ? (k0 + 4) : k0;           // last iter: reload (discarded)
        int ka = k1 + 2 * half;
        // issue next-step fragments first ...
        v2f a_nxt = *(const v2f*)(Ab + k1);
        v2f b_nxt[NT];
        #pragma unroll
        for (int t = 0; t < NT; ++t) {
            if (BMODE == 1) b_nxt[t] = *(const v2f*)(Bb[t] + ka);
            else { b_nxt[t][0] = Bb[t][(long long)ka * ldbK];
                   b_nxt[t][1] = Bb[t][(long long)(ka + 1) * ldbK]; }
        }
        // ... then consume current-step fragments
        #pragma unroll
        for (int t = 0; t < NT; ++t)
            acc[t] = __builtin_amdgcn_wmma_f32_16x16x4_f32(
                false, a_cur, false, b_cur[t], (short)0, acc[t], false, false);
        a_cur = a_nxt;
        #pragma unroll
        for (int t = 0; t < NT; ++t) b_cur[t] = b_nxt[t];
    }

    #pragma unroll
    for (int t = 0; t < NT; ++t) {
        int n = n0 + t * 16 + l;
        #pragma unroll
        for (int r = 0; r < 8; ++r) {
            int m = m0 + half * 8 + r;
            if (m < M && n < N) {
                float v = acc[t][r];
                if (bias)  v += bias[n];
                if (addIn) v += addIn[(long long)m * ldc + n];
                if (act == 1)      v = fmaxf(v, 0.f);
                else if (act == 2) v = 1.f / (1.f + __expf(-v));
                else if (act == 3) v = fminf(fmaxf(v, -10.f), 10.f);
                C[(long long)m * ldc + n] = v;
            }
        }
    }
}

// ---------------------------------------------------------------------------
// LayerNorm over width 256, one wave per row. out = LN(x (+ res)) * g + b
// ---------------------------------------------------------------------------
__global__ void layernorm_kernel(const float* __restrict__ x, const float* __restrict__ res,
                                 const float* __restrict__ g, const float* __restrict__ beta,
                                 float* __restrict__ out)
{
    int row  = blockIdx.x;
    int lane = threadIdx.x;
    const float* xr = x + (long long)row * 256;
    const float* rr = res ? res + (long long)row * 256 : nullptr;
    float v[8];
    float s = 0.f;
    #pragma unroll
    for (int i = 0; i < 8; ++i) {
        float t = xr[lane + 32 * i];
        if (rr) t += rr[lane + 32 * i];
        v[i] = t; s += t;
    }
    #pragma unroll
    for (int o = 16; o > 0; o >>= 1) s += __shfl_xor(s, o, 32);
    float mean = s * (1.f / 256.f);
    float var = 0.f;
    #pragma unroll
    for (int i = 0; i < 8; ++i) { float d = v[i] - mean; var += d * d; }
    #pragma unroll
    for (int o = 16; o > 0; o >>= 1) var += __shfl_xor(var, o, 32);
    var *= (1.f / 256.f);
    float inv = rsqrtf(var + 1e-5f);
    #pragma unroll
    for (int i = 0; i < 8; ++i) {
        int c = lane + 32 * i;
        out[(long long)row * 256 + c] = (v[i] - mean) * inv * g[c] + beta[c];
    }
}

// ---------------------------------------------------------------------------
// Row softmax (width <= 320), one wave per row. out = softmax(in * scale)
// ---------------------------------------------------------------------------
__global__ void softmax_rows_kernel(const float* __restrict__ in, float* __restrict__ out,
                                    int width, float scale)
{
    int row  = blockIdx.x;
    int lane = threadIdx.x;
    const float* xr = in + (long long)row * width;
    float e[10];
    int cnt = 0;
    float m = -3.4e38f;
    for (int j = lane; j < width; j += 32) { float t = xr[j] * scale; e[cnt++] = t; m = fmaxf(m, t); }
    #pragma unroll
    for (int o = 16; o > 0; o >>= 1) m = fmaxf(m, __shfl_xor(m, o, 32));
    float s = 0.f;
    for (int c = 0; c < cnt; ++c) { e[c] = __expf(e[c] - m); s += e[c]; }
    #pragma unroll
    for (int o = 16; o > 0; o >>= 1) s += __shfl_xor(s, o, 32);
    float invs = 1.f / s;
    cnt = 0;
    for (int j = lane; j < width; j += 32) out[(long long)row * width + j] = e[cnt++] * invs;
}

// ---------------------------------------------------------------------------
// Deformable attention sampling: thread per (b,q,h,d). Samples directly from
// memory (b, 8500, 256): v[b,h,d,l] = memory[b, l, h*32+d].
// ---------------------------------------------------------------------------
__device__ __forceinline__ float ds_fetch(const float* memb, int start, int W, int H,
                                          int x, int y, int cOff)
{
    if (x < 0 || x >= W || y < 0 || y >= H) return 0.f;
    return memb[(long long)(start + y * W + x) * 256 + cOff];
}

__global__ void deform_sample_kernel(const float* __restrict__ mem, const float* __restrict__ off,
                                     const float* __restrict__ aw, const float* __restrict__ refD,
                                     float* __restrict__ t2)
{
    int tid = blockIdx.x * blockDim.x + threadIdx.x;
    if (tid >= BB * NQ * NH * HD) return;
    int d  = tid & 31;
    int h  = (tid >> 5) & 7;
    int q  = (tid >> 8) % NQ;
    int b  = tid / (NQ * 256);

    const float* r = refD + ((long long)b * NQ + q) * 4;
    float cx = r[0], cy = r[1], rw = r[2], rh = r[3];
    const float* ofp = off + (((long long)b * NQ + q) * NH + h) * 30;
    const float* awp = aw  + (((long long)b * NQ + q) * NH + h) * TP;
    const float* memb = mem + (long long)b * LTOT * 256;
    int cOff = h * 32 + d;

    float acc = 0.f;
    #pragma unroll
    for (int p = 0; p < TP; ++p) {
        int lvl = (p < 3) ? 0 : (p < 9) ? 1 : (p < 12) ? 2 : 3;
        int H = (lvl == 0) ? 80 : (lvl == 1) ? 40 : (lvl == 2) ? 20 : 10;
        int W = H;
        int start = (lvl == 0) ? 0 : (lvl == 1) ? 6400 : (lvl == 2) ? 8000 : 8400;
        float nps = (lvl == 1) ? (1.f / 6.f) : (1.f / 3.f);

        float lx = cx + ofp[2 * p]     * nps * rw * 0.5f;
        float ly = cy + ofp[2 * p + 1] * nps * rh * 0.5f;
        float xf = lx * W - 0.5f;
        float yf = ly * H - 0.5f;
        float x0f = floorf(xf), y0f = floorf(yf);
        float wx = xf - x0f, wy = yf - y0f;
        int x0 = (int)x0f, y0 = (int)y0f;

        float g00 = ds_fetch(memb, start, W, H, x0,     y0,     cOff);
        float g01 = ds_fetch(memb, start, W, H, x0 + 1, y0,     cOff);
        float g10 = ds_fetch(memb, start, W, H, x0,     y0 + 1, cOff);
        float g11 = ds_fetch(memb, start, W, H, x0 + 1, y0 + 1, cOff);
        float sampled = g00 * (1.f - wx) * (1.f - wy) + g01 * wx * (1.f - wy)
                      + g10 * (1.f - wx) * wy         + g11 * wx * wy;
        acc += awp[p] * sampled;
    }
    t2[tid] = acc;
}

// ------------------------- elementwise helpers -----------------------------
__global__ void add2_kernel(const float* __restrict__ a, const float* __restrict__ b,
                            float* __restrict__ o, int n)
{ int i = blockIdx.x * blockDim.x + threadIdx.x; if (i < n) o[i] = a[i] + b[i]; }

__global__ void copy_kernel(const float* __restrict__ a, float* __restrict__ o, int n)
{ int i = blockIdx.x * blockDim.x + threadIdx.x; if (i < n) o[i] = a[i]; }

__global__ void sigmoid_kernel(const float* __restrict__ a, float* __restrict__ o, int n)
{ int i = blockIdx.x * blockDim.x + threadIdx.x; if (i < n) o[i] = 1.f / (1.f + __expf(-a[i])); }

__global__ void cat_kernel(const float* __restrict__ a, const float* __restrict__ b,
                           float* __restrict__ o, int n)  // rows of 512 = [a(256) | b(256)]
{
    int i = blockIdx.x * blockDim.x + threadIdx.x;
    if (i >= n) return;
    int row = i >> 9, c = i & 511;
    o[i] = (c < 256) ? a[(long long)row * 256 + c] : b[(long long)row * 256 + (c - 256)];
}

__global__ void gate_combine_kernel(const float* __restrict__ gates, const float* __restrict__ a,
                                    const float* __restrict__ b, float* __restrict__ o, int n)
{
    int i = blockIdx.x * blockDim.x + threadIdx.x;
    if (i >= n) return;
    int row = i >> 8, c = i & 255;
    float g1 = gates[(long long)row * 512 + c];
    float g2 = gates[(long long)row * 512 + 256 + c];
    o[i] = g1 * a[i] + g2 * b[i];
}

__global__ void pre_bbox_kernel(const float* __restrict__ raw, const float* __restrict__ refD,
                                float* __restrict__ initRef, float* __restrict__ dout)
{
    int i = blockIdx.x * blockDim.x + threadIdx.x;
    if (i >= ROWS * 4) return;
    float s = refD[i];
    s = fminf(fmaxf(s, 0.f), 1.f);
    float inv = logf(fmaxf(s, 1e-5f) / fmaxf(1.f - s, 1e-5f));
    float v = 1.f / (1.f + __expf(-(raw[i] + inv)));
    initRef[i] = v;
    dout[O_PREB + i] = v;
    #pragma unroll
    for (int l = 0; l < 6; ++l) dout[O_REFS + l * (ROWS * 4) + i] = v;
}

__global__ void integral_kernel(const float* __restrict__ prob, const float* __restrict__ w,
                                float* __restrict__ dist)
{
    int i = blockIdx.x * blockDim.x + threadIdx.x;   // (b*300+q)*4 + j
    if (i >= ROWS * 4) return;
    const float* pr = prob + (long long)i * REGBINS;
    float s = 0.f;
    #pragma unroll
    for (int t = 0; t < REGBINS; ++t) s += pr[t] * w[t];
    dist[i] = s;
}

__global__ void dist2bbox_kernel(const float* __restrict__ initRef, const float* __restrict__ dist,
                                 float* __restrict__ refD, float* __restrict__ doutBbox)
{
    int i = blockIdx.x * blockDim.x + threadIdx.x;
    if (i >= ROWS) return;
    const float* p = initRef + (long long)i * 4;
    const float* d = dist + (long long)i * 4;
    float hw = p[2] * 0.25f, hh = p[3] * 0.25f;   // wh / reg_scale, reg_scale = 4
    float x1 = p[0] - (2.f + d[0]) * hw;
    float y1 = p[1] - (2.f + d[1]) * hh;
    float x2 = p[0] + (2.f + d[2]) * hw;
    float y2 = p[1] + (2.f + d[3]) * hh;
    float o0 = (x1 + x2) * 0.5f, o1 = (y1 + y2) * 0.5f, o2 = x2 - x1, o3 = y2 - y1;
    float* ob = doutBbox + (long long)i * 4;
    ob[0] = o0; ob[1] = o1; ob[2] = o2; ob[3] = o3;
    float* r = refD + (long long)i * 4;
    r[0] = o0; r[1] = o1; r[2] = o2; r[3] = o3;
}

__global__ void topk_stat_kernel(const float* __restrict__ prob, float* __restrict__ stat)
{
    int i = blockIdx.x * blockDim.x + threadIdx.x;   // (b*300+q)*4 + j
    if (i >= ROWS * 4) return;
    float v[REGBINS];
    const float* pr = prob + (long long)i * REGBINS;
    #pragma unroll
    for (int t = 0; t < REGBINS; ++t) v[t] = pr[t];
    float tk[4];
    #pragma unroll
    for (int s = 0; s < 4; ++s) {
        int mi = 0; float mv = v[0];
        for (int t = 1; t < REGBINS; ++t) if (v[t] > mv) { mv = v[t]; mi = t; }
        tk[s] = mv; v[mi] = -1e30f;
    }
    int bq = i >> 2, j = i & 3;
    float* st = stat + (long long)bq * 20 + j * 5;
    st[0] = tk[0]; st[1] = tk[1]; st[2] = tk[2]; st[3] = tk[3];
    st[4] = (tk[0] + tk[1] + tk[2] + tk[3]) * 0.25f;
}

__global__ void lqe_add_kernel(const float* __restrict__ scores, const float* __restrict__ qsc,
                               float* __restrict__ out, int n)
{
    int i = blockIdx.x * blockDim.x + threadIdx.x;
    if (i >= n) return;
    out[i] = scores[i] + qsc[i / NCLS];
}

// ============================================================================
// Host orchestration
// ============================================================================
static void launch_gemm_full(hipStream_t s,
                             const float* A, long long sA1, long long sA2, int lda,
                             const float* B, long long sB1, long long sB2, int ldbK, int ldbN,
                             const float* bias, const float* addIn,
                             float* C, long long sC1, long long sC2, int ldc,
                             int M, int N, int K, int batch, int inner, int act)
{
    int my = (M + 15) / 16;
    if (N % 64 == 0) {
        dim3 g(N / 64, my, batch);
        if (ldbK == 1) gemm_wmma_f32<4, 1><<<g, dim3(32), 0, s>>>(A, sA1, sA2, lda, B, sB1, sB2, ldbK, ldbN, bias, addIn, C, sC1, sC2, ldc, M, N, K, inner, act);
        else           gemm_wmma_f32<4, 0><<<g, dim3(32), 0, s>>>(A, sA1, sA2, lda, B, sB1, sB2, ldbK, ldbN, bias, addIn, C, sC1, sC2, ldc, M, N, K, inner, act);
    } else if (N % 32 == 0) {
        dim3 g(N / 32, my, batch);
        if (ldbK == 1) gemm_wmma_f32<2, 1><<<g, dim3(32), 0, s>>>(A, sA1, sA2, lda, B, sB1, sB2, ldbK, ldbN, bias, addIn, C, sC1, sC2, ldc, M, N, K, inner, act);
        else           gemm_wmma_f32<2, 0><<<g, dim3(32), 0, s>>>(A, sA1, sA2, lda, B, sB1, sB2, ldbK, ldbN, bias, addIn, C, sC1, sC2, ldc, M, N, K, inner, act);
    } else {
        dim3 g((N + 15) / 16, my, batch);
        if (ldbK == 1) gemm_wmma_f32<1, 1><<<g, dim3(32), 0, s>>>(A, sA1, sA2, lda, B, sB1, sB2, ldbK, ldbN, bias, addIn, C, sC1, sC2, ldc, M, N, K, inner, act);
        else           gemm_wmma_f32<1, 0><<<g, dim3(32), 0, s>>>(A, sA1, sA2, lda, B, sB1, sB2, ldbK, ldbN, bias, addIn, C, sC1, sC2, ldc, M, N, K, inner, act);
    }
}

static inline void gemm(hipStream_t s, const float* A, const float* W, const float* bias,
                        float* C, int M, int N, int K, int act, const float* addIn = nullptr)
{
    launch_gemm_full(s, A, 0, 0, K, W, 0, 0, N, 1, bias, addIn, C, 0, 0, N, M, N, K, 1, 1, act);
}

#define EW(kernel, n, ...) kernel<<<((n) + 255) / 256, 256, 0, stream>>>(__VA_ARGS__)

extern "C" void kernel_launch(void* const* d_in, const int* in_sizes, int n_in,
                              void* d_out_v, int out_size, void* d_ws, size_t ws_size,
                              hipStream_t stream)
{
    (void)in_sizes; (void)n_in; (void)out_size; (void)ws_size;
    float* d_out = (float*)d_out_v;

    const float* target  = (const float*)d_in[0];
    const float* refUn   = (const float*)d_in[1];
    const float* memory  = (const float*)d_in[2];

    int pi = 3;
    auto P = [&]() { return (const float*)d_in[pi++]; };

    // ---- params (JAX pytree order, see header) ----
    const float *bb_b[6][3], *bb_w[6][3];
    for (int i = 0; i < 6; ++i) for (int j = 0; j < 3; ++j) { bb_b[i][j] = P(); bb_w[i][j] = P(); }
    const float* wdist = P();
    struct Layer {
        const float *aw_b, *aw_w, *off_b, *off_w, *ff1_b, *ff1_w, *ff2_b, *ff2_w,
                    *gate_b, *gate_w, *gn_g, *gn_b, *n1_g, *n1_b, *n3_g, *n3_b,
                    *k_b, *k_w, *o_b, *o_w, *q_b, *q_w, *v_b, *v_w;
    } Ls[6];
    for (int i = 0; i < 6; ++i) {
        Layer& L = Ls[i];
        L.aw_b = P();  L.aw_w = P();  L.off_b = P(); L.off_w = P();
        L.ff1_b = P(); L.ff1_w = P(); L.ff2_b = P(); L.ff2_w = P();
        L.gate_b = P(); L.gate_w = P();
        L.gn_g = P(); L.gn_b = P();
        L.n1_g = P(); L.n1_b = P();
        L.n3_g = P(); L.n3_b = P();
        L.k_b = P(); L.k_w = P(); L.o_b = P(); L.o_w = P();
        L.q_b = P(); L.q_w = P(); L.v_b = P(); L.v_w = P();
    }
    const float *lqe_b[6][2], *lqe_w[6][2];
    for (int i = 0; i < 6; ++i) for (int j = 0; j < 2; ++j) { lqe_b[i][j] = P(); lqe_w[i][j] = P(); }
    const float *pb_b[3], *pb_w[3];
    for (int j = 0; j < 3; ++j) { pb_b[j] = P(); pb_w[j] = P(); }
    const float *qp_b[2], *qp_w[2];
    for (int j = 0; j < 2; ++j) { qp_b[j] = P(); qp_w[j] = P(); }
    const float *sc_b[6], *sc_w[6];
    for (int i = 0; i < 6; ++i) { sc_b[i] = P(); sc_w[i] = P(); }

    // ---- workspace carve (~173 MB fp32) ----
    char* wp = (char*)d_ws;
    auto alloc = [&](size_t n) { float* p = (float*)wp; wp += n * sizeof(float); return p; };
    float* cur     = alloc(ROWS * 256);
    float* prevOut = alloc(ROWS * 256);
    float* addbuf  = alloc(ROWS * 256);
    float* qpeh    = alloc(ROWS * 512);
    float* qpe     = alloc(ROWS * 256);
    float* qbuf    = alloc(ROWS * 256);
    float* kbuf    = alloc(ROWS * 256);
    float* vbuf    = alloc(ROWS * 256);
    float* att     = alloc((size_t)BB * NH * NQ * NQ);   // 11.52M
    float* saout   = alloc(ROWS * 256);
    float* t256    = alloc(ROWS * 256);
    float* offb    = alloc(ROWS * NH * TP * 2);
    float* awb     = alloc(ROWS * NH * TP + 2);          // keep 8B alignment
    float* t2      = alloc(ROWS * 256);
    float* cat     = alloc(ROWS * 512);
    float* gates   = alloc(ROWS * 512);
    float* ffh     = alloc(ROWS * DFF);
    float* h1      = alloc(ROWS * 256);
    float* h2      = alloc(ROWS * 256);
    float* cornA   = alloc(ROWS * 4 * REGBINS);
    float* cornB   = alloc(ROWS * 4 * REGBINS);
    float* prob    = alloc(ROWS * 4 * REGBINS);
    float* dist    = alloc(ROWS * 4);
    float* refD    = alloc(ROWS * 4);
    float* initRef = alloc(ROWS * 4);
    float* pbraw   = alloc(ROWS * 4);
    float* scoresb = alloc(ROWS * NCLS);
    float* stat    = alloc(ROWS * 20);
    float* l1      = alloc(ROWS * 64);
    float* qsc     = alloc(ROWS);

    // ---- init ----
    EW(copy_kernel, ROWS * 256, target, cur, ROWS * 256);
    EW(sigmoid_kernel, ROWS * 4, refUn, refD, ROWS * 4);

    const float ATT_SCALE = 0.17677669529663688f;  // 1/sqrt(32)

    for (int i = 0; i < 6; ++i) {
        const Layer& L = Ls[i];
        float* cornCur  = (i & 1) ? cornB : cornA;
        float* cornPrev = (i & 1) ? cornA : cornB;

        // query pos embed: mlp(refD): 4->512 relu, 512->256, clip(+-10)
        gemm(stream, refD, qp_w[0], qp_b[0], qpeh, ROWS, 512, 4, 1);
        gemm(stream, qpeh, qp_w[1], qp_b[1], qpe, ROWS, 256, 512, 3);

        // ---- self attention ----
        EW(add2_kernel, ROWS * 256, cur, qpe, addbuf, ROWS * 256);
        gemm(stream, addbuf, L.q_w, L.q_b, qbuf, ROWS, 256, 256, 0);
        gemm(stream, addbuf, L.k_w, L.k_b, kbuf, ROWS, 256, 256, 0);
        gemm(stream, cur,    L.v_w, L.v_b, vbuf, ROWS, 256, 256, 0);

        // scores[b,h,q,k] = q . k   (batched over 128 (b,h) pairs; B has ldbK=1)
        launch_gemm_full(stream,
            qbuf, (long long)NQ * 256, 32, 256,
            kbuf, (long long)NQ * 256, 32, 1, 256,
            nullptr, nullptr,
            att, (long long)NH * NQ * NQ, (long long)NQ * NQ, NQ,
            NQ, NQ, HD, BB * NH, NH, 0);
        softmax_rows_kernel<<<BB * NH * NQ, 32, 0, stream>>>(att, att, NQ, ATT_SCALE);
        // att @ V -> saout[b,q,h*32+d]
        launch_gemm_full(stream,
            att, (long long)NH * NQ * NQ, (long long)NQ * NQ, NQ,
            vbuf, (long long)NQ * 256, 32, 256, 1,
            nullptr, nullptr,
            saout, (long long)NQ * 256, 32, 256,
            NQ, HD, NQ, BB * NH, NH, 0);
        gemm(stream, saout, L.o_w, L.o_b, t256, ROWS, 256, 256, 0);
        layernorm_kernel<<<ROWS, 32, 0, stream>>>(cur, t256, L.n1_g, L.n1_b, cur);

        // ---- deformable cross attention ----
        EW(add2_kernel, ROWS * 256, cur, qpe, addbuf, ROWS * 256);
        gemm(stream, addbuf, L.off_w, L.off_b, offb, ROWS, NH * TP * 2, 256, 0);
        gemm(stream, addbuf, L.aw_w,  L.aw_b,  awb,  ROWS, NH * TP,     256, 0);
        softmax_rows_kernel<<<ROWS * NH, 32, 0, stream>>>(awb, awb, TP, 1.f);
        EW(deform_sample_kernel, ROWS * 256, memory, offb, awb, refD, t2);

        // gated fusion
        EW(cat_kernel, ROWS * 512, cur, t2, cat, ROWS * 512);
        gemm(stream, cat, L.gate_w, L.gate_b, gates, ROWS, 512, 512, 2);
        EW(gate_combine_kernel, ROWS * 256, gates, cur, t2, t256, ROWS * 256);
        layernorm_kernel<<<ROWS, 32, 0, stream>>>(t256, nullptr, L.gn_g, L.gn_b, cur);

        // FFN
        gemm(stream, cur, L.ff1_w, L.ff1_b, ffh, ROWS, DFF, 256, 1);
        gemm(stream, ffh, L.ff2_w, L.ff2_b, t256, ROWS, 256, DFF, 0);
        layernorm_kernel<<<ROWS, 32, 0, stream>>>(cur, t256, L.n3_g, L.n3_b, cur);

        // ---- heads ----
        if (i == 0) {
            gemm(stream, cur, pb_w[0], pb_b[0], h1, ROWS, 256, 256, 1);
            gemm(stream, h1,  pb_w[1], pb_b[1], h2, ROWS, 256, 256, 1);
            gemm(stream, h2,  pb_w[2], pb_b[2], pbraw, ROWS, 4, 256, 0);
            EW(pre_bbox_kernel, ROWS * 4, pbraw, refD, initRef, d_out);
        }

        // bbox head: mlp(cur + prevOut) + prevCorners
        const float* bbin = cur;
        if (i > 0) { EW(add2_kernel, ROWS * 256, cur, prevOut, addbuf, ROWS * 256); bbin = addbuf; }
        gemm(stream, bbin, bb_w[i][0], bb_b[i][0], h1, ROWS, 256, 256, 1);
        gemm(stream, h1,   bb_w[i][1], bb_b[i][1], h2, ROWS, 256, 256, 1);
        gemm(stream, h2,   bb_w[i][2], bb_b[i][2], cornCur, ROWS, 4 * REGBINS, 256, 0,
             (i > 0) ? cornPrev : nullptr);

        // integral + distance2bbox
        softmax_rows_kernel<<<ROWS * 4, 32, 0, stream>>>(cornCur, prob, REGBINS, 1.f);
        EW(integral_kernel, ROWS * 4, prob, wdist, dist);
        EW(dist2bbox_kernel, ROWS, initRef, dist, refD, d_out + O_BBOX + (size_t)i * ROWS * 4);

        // score head + LQE
        gemm(stream, cur, sc_w[i], sc_b[i], scoresb, ROWS, NCLS, 256, 0);
        if (i == 0) EW(copy_kernel, ROWS * NCLS, scoresb, d_out + O_PRES, ROWS * NCLS);
        EW(topk_stat_kernel, ROWS * 4, prob, stat);
        gemm(stream, stat, lqe_w[i][0], lqe_b[i][0], l1, ROWS, 64, 20, 1);
        gemm(stream, l1,   lqe_w[i][1], lqe_b[i][1], qsc, ROWS, 1, 64, 0);
        EW(lqe_add_kernel, ROWS * NCLS, scoresb, qsc, d_out + O_LOGIT + (size_t)i * ROWS * NCLS,
           ROWS * NCLS);

        // export corners, save output_detach
        EW(copy_kernel, ROWS * 4 * REGBINS, cornCur, d_out + O_CORN + (size_t)i * ROWS * 4 * REGBINS,
           ROWS * 4 * REGBINS);
        EW(copy_kernel, ROWS * 256, cur, prevOut, ROWS * 256);
    }
}